// SingleQubitRegressor_9483287790319
// MI455X (gfx1250) — compile-verified
//
#include <hip/hip_runtime.h>
#include <stdint.h>
#include <stddef.h>

// ---------------------------------------------------------------------------
// Single-qubit data-reuploading circuit, Bloch-vector formulation.
//   Per step t:  Bloch rotate about X by x[b,t], about Y by theta[t,0],
//                about Z by theta[t,1].  Output = Z component * w + b.
// Data path: global -> LDS via CDNA5 async DMA (global_load_async_to_lds_b128,
// drained with s_wait_asynccnt), padded-row LDS layout, ds_load_b128 row pull,
// fully unrolled 12-FMA/step VALU loop with SGPR-resident theta trig.
// ---------------------------------------------------------------------------

#define W24   24
#define TPB   256
#define ROWF  28            // padded row stride in floats (112 B, 16B aligned)
#define F4_PER_ROW 6        // 24 floats = 6 float4
#define F4_PER_TILE (TPB * F4_PER_ROW)   // 1536 float4 per block tile

#define INV_2PI 0.15915494309189535f

// Tiny kernel: precompute cos/sin of the 24x2 shared angles into workspace.
// |theta| ~ 0.01*N(0,1) -> native v_sin/v_cos are exact to ~1e-7 relative
// here; keeps this serial graph node as short as possible.
// Layout: trig[4t+0]=cos(th0), [4t+1]=sin(th0), [4t+2]=cos(th1), [4t+3]=sin(th1)
__global__ void sq_trig_kernel(const float* __restrict__ theta,
                               float* __restrict__ trig) {
  int t = threadIdx.x;
  if (t < W24) {
    float ub = theta[2 * t + 0] * INV_2PI;
    float uc = theta[2 * t + 1] * INV_2PI;
    trig[4 * t + 0] = __builtin_amdgcn_cosf(ub);
    trig[4 * t + 1] = __builtin_amdgcn_sinf(ub);
    trig[4 * t + 2] = __builtin_amdgcn_cosf(uc);
    trig[4 * t + 3] = __builtin_amdgcn_sinf(uc);
  }
}

__global__ __launch_bounds__(TPB) void sq_main_kernel(
    const float* __restrict__ x, const float* __restrict__ trig,
    const float* __restrict__ rw, const float* __restrict__ rb,
    float* __restrict__ out, int nB) {
  // Padded tile: 256 rows x 28 floats (row = one window's 24 samples + pad)
  __shared__ __align__(16) float smem[TPB * ROWF];

  const int tid = threadIdx.x;
  const long long tile4 = (long long)blockIdx.x * F4_PER_TILE;
  const long long tot4  = (long long)nB * F4_PER_ROW;
  const unsigned lbase  = (unsigned)(size_t)(&smem[0]);   // LDS byte address
  const unsigned long long xbase = (unsigned long long)(size_t)x;

  // --- Async DMA: global tile -> padded LDS rows (CDNA5 async-to-LDS path) --
#pragma unroll
  for (int k = 0; k < F4_PER_ROW; ++k) {
    int l4 = tid + k * TPB;                 // float4 index within tile
    int r  = l4 / F4_PER_ROW;               // destination row (window)
    int c  = l4 - r * F4_PER_ROW;           // float4 column within row
    long long g4 = tile4 + l4;
    if (g4 >= tot4) g4 = tot4 - 1;          // clamp tail (harmless dup read)
    unsigned lds = lbase + (unsigned)(r * (ROWF * 4) + c * 16);
    unsigned long long ga = xbase + (unsigned long long)g4 * 16ull;
    asm volatile("global_load_async_to_lds_b128 %0, %1, off"
                 :: "v"(lds), "v"(ga)
                 : "memory");
  }
  asm volatile("s_wait_asynccnt 0x0" ::: "memory");
  __syncthreads();

  // --- Pull this lane's row into registers (6x ds_load_b128) ---------------
  float4 rowv[F4_PER_ROW];
  const float4* rp = (const float4*)(&smem[tid * ROWF]);
#pragma unroll
  for (int k = 0; k < F4_PER_ROW; ++k) rowv[k] = rp[k];
  const float* row = (const float*)rowv;

  // --- Bloch-vector scan: start at |0> = (0,0,1) ---------------------------
  float X = 0.0f, Y = 0.0f, Z = 1.0f;
#pragma unroll
  for (int t = 0; t < W24; ++t) {
    // One shared range-reduction multiply feeds both v_sin_f32 and v_cos_f32
    // (TRANS pipe, co-executes with the FMA stream).
    float u = row[t] * INV_2PI;
    float sa = __builtin_amdgcn_sinf(u);
    float ca = __builtin_amdgcn_cosf(u);
    // rotate about X by x_t: (y,z)  -- independent pair, VOPD-friendly
    float y = fmaf(ca, Y, -sa * Z);
    float z = fmaf(sa, Y,  ca * Z);
    // rotate about Y by theta[t,0]: (z,x)   [uniform -> SGPR operands]
    float cb = trig[4 * t + 0], sb = trig[4 * t + 1];
    float xn = fmaf(cb, X,  sb * z);
    float zn = fmaf(cb, z, -sb * X);
    // rotate about Z by theta[t,1]: (x,y)
    float cc = trig[4 * t + 2], sc = trig[4 * t + 3];
    X = fmaf(cc, xn, -sc * y);
    Y = fmaf(sc, xn,  cc * y);
    Z = zn;
  }

  int w = blockIdx.x * TPB + tid;
  if (w < nB) out[w] = fmaf(Z, rw[0], rb[0]);   // <Z> * readout_w + readout_b
}

extern "C" void kernel_launch(void* const* d_in, const int* in_sizes, int n_in,
                              void* d_out, int out_size, void* d_ws, size_t ws_size,
                              hipStream_t stream) {
  const float* x     = (const float*)d_in[0];   // [B, 24] f32
  const float* theta = (const float*)d_in[1];   // [24, 2] f32
  const float* rw    = (const float*)d_in[2];   // [1, 1]  f32
  const float* rb    = (const float*)d_in[3];   // [1]     f32
  float* out  = (float*)d_out;
  float* trig = (float*)d_ws;                   // 96 floats of precomputed trig

  int nB = in_sizes[0] / W24;

  sq_trig_kernel<<<1, 32, 0, stream>>>(theta, trig);

  int blocks = (nB + TPB - 1) / TPB;
  sq_main_kernel<<<blocks, TPB, 0, stream>>>(x, trig, rw, rb, out, nB);
}